// BilinearRoiPooling_89962384982483
// MI455X (gfx1250) — compile-verified
//
#include <hip/hip_runtime.h>
#include <stdint.h>

typedef __attribute__((ext_vector_type(2))) float v2f;
typedef __attribute__((ext_vector_type(8))) float v8f;

#define NC   512          // channels
#define FH   64           // feature height
#define FW   256          // feature width
#define FHW  (FH * FW)
#define NP   49           // 7x7 output positions
#define NK   196          // 14x14 tap grid
#define CCHUNK 64         // channels staged in LDS per pass
#define FSTR 197          // sF row stride (floats); odd -> conflict-free A reads
#define WSTR 72           // sWd row stride; 2*72 mod 64 == 16 -> disjoint banks for the
                          // two K-rows read simultaneously by lanes 0-15 / 16-31

__device__ __forceinline__ unsigned lds_addr32(const void* p) {
    // generic -> LDS address space is low-32-bit truncation on AMDGPU
    return (unsigned)(uintptr_t)p;
}

// One 16(ch) x 16(pos) D-tile: K=196 in chunks of 4, statically pruned to the
// chunks whose taps feed positions [NBASE, NBASE+16). Fh/Wh are per-lane base
// pointers already offset by 2*hi (lanes 16-31 hold K+2 per the ISA layout).
template<int NBASE>
__device__ __forceinline__ v8f tile_gemm(const float* __restrict__ Fh,
                                         const float* __restrict__ Wh) {
    v8f acc = {};
#pragma unroll
    for (int kc = 0; kc < 49; ++kc) {
        int pmn = 63, pmx = 0;
#pragma unroll
        for (int q = 0; q < 4; ++q) {
            const int k = 4 * kc + q;
            const int p = ((k / 14) >> 1) * 7 + ((k % 14) >> 1);
            pmn = p < pmn ? p : pmn;
            pmx = p > pmx ? p : pmx;
        }
        if (pmx < NBASE || pmn > NBASE + 15) continue;  // constant-folds after unroll
        v2f af, bf;
        af.x = Fh[4 * kc];                  // ds_load, constant offset
        af.y = Fh[4 * kc + 1];
        bf.x = Wh[(4 * kc) * WSTR];         // ds_load, constant offset (<64KB)
        bf.y = Wh[(4 * kc + 1) * WSTR];
        acc = __builtin_amdgcn_wmma_f32_16x16x4_f32(
            false, af, false, bf, (short)0, acc, false, false);
    }
    return acc;
}

__launch_bounds__(256)
__global__ void roi_bilinear_wmma_kernel(const float* __restrict__ feats,
                                         const float* __restrict__ boxes,
                                         const int*   __restrict__ imHp,
                                         const int*   __restrict__ imWp,
                                         float*       __restrict__ out)
{
    // sWd first so its ds offsets stay under the 16-bit immediate limit
    __shared__ float sWd[NK * WSTR];        // 56,448 B dense weight matrix
    __shared__ float sF[CCHUNK * FSTR];     // 50,432 B gathered taps
    __shared__ int   s_xs[14], s_ys[14];
    __shared__ float s_wx[14], s_wy[14];

    const int b   = blockIdx.x;
    const int tid = threadIdx.x;

    // ---- per-box separable sampling grid ----
    if (tid < 7) { // x side, ow = tid
        const float xc  = boxes[4 * b + 0];
        const float bw  = boxes[4 * b + 2];
        const float Wim = (float)imWp[0];
        const float a   = (bw - 1.0f) / (Wim - 1.0f);
        const float t   = (2.0f * xc - Wim - 1.0f) / (Wim - 1.0f);
        const float g   = -1.0f + (2.0f / 6.0f) * (float)tid;
        float ix = (a * g + t + 1.0f) * 0.5f * (float)(FW - 1);
        ix = fminf(fmaxf(ix, 0.0f), (float)(FW - 1));
        const float x0 = floorf(ix);
        const float x1 = fminf(x0 + 1.0f, (float)(FW - 1));
        const float wx = ix - x0;
        s_xs[2 * tid]     = (int)x0;
        s_xs[2 * tid + 1] = (int)x1;
        s_wx[2 * tid]     = 1.0f - wx;
        s_wx[2 * tid + 1] = wx;
    } else if (tid >= 8 && tid < 15) { // y side, oh = tid - 8
        const int   oh  = tid - 8;
        const float yc  = boxes[4 * b + 1];
        const float bh  = boxes[4 * b + 3];
        const float Him = (float)imHp[0];
        const float a   = (bh - 1.0f) / (Him - 1.0f);
        const float t   = (2.0f * yc - Him - 1.0f) / (Him - 1.0f);
        const float g   = -1.0f + (2.0f / 6.0f) * (float)oh;
        float iy = (a * g + t + 1.0f) * 0.5f * (float)(FH - 1);
        iy = fminf(fmaxf(iy, 0.0f), (float)(FH - 1));
        const float y0 = floorf(iy);
        const float y1 = fminf(y0 + 1.0f, (float)(FH - 1));
        const float wy = iy - y0;
        s_ys[2 * oh]     = (int)y0;
        s_ys[2 * oh + 1] = (int)y1;
        s_wy[2 * oh]     = 1.0f - wy;
        s_wy[2 * oh + 1] = wy;
    }
    __syncthreads();

    // ---- build dense K x 64 weight matrix (one nonzero per K-row) ----
    for (int e = tid; e < NK * 64; e += 256) {
        const int k = e >> 6;
        const int p = e & 63;
        const int j = k / 14, i = k - 14 * j;
        const int pk = (j >> 1) * 7 + (i >> 1);
        sWd[k * WSTR + p] = (pk == p) ? s_wy[j] * s_wx[i] : 0.0f;
    }
    // first in-loop barrier covers sWd visibility

    const int wv   = __builtin_amdgcn_readfirstlane((int)(tid >> 5)); // scalar wave id
    const int lane = tid & 31;
    const int m    = lane & 15;   // M row (channels) / N col (positions)
    const int hi   = lane >> 4;   // upper lane group holds K+2
    const int ctile = wv >> 1;    // 4 channel tiles per chunk, 2 waves each
    const int nsel  = wv & 1;     // which pair of N-tiles this wave owns

    for (int cbase = 0; cbase < NC; cbase += CCHUNK) {
        // async gather: 64 ch x 196 taps, global -> LDS via CDNA5 async copy
        for (int e = tid; e < CCHUNK * NK; e += 256) {   // 12544 = 49 * 256, no tail
            const int c = e / NK;
            const int k = e - c * NK;
            const int j = k / 14, i = k - 14 * j;
            const float* gsrc =
                feats + (size_t)(cbase + c) * FHW + s_ys[j] * FW + s_xs[i];
            const unsigned ldst = lds_addr32(&sF[c * FSTR + k]);
            asm volatile("global_load_async_to_lds_b32 %0, %1, off"
                         :: "v"(ldst), "v"((uint64_t)(uintptr_t)gsrc)
                         : "memory");
        }
        asm volatile("s_wait_asynccnt 0x0" ::: "memory");
        __syncthreads();

        const float* Fh  = &sF[(ctile * 16 + m) * FSTR + 2 * hi];
        const float* Wh  = &sWd[(2 * hi) * WSTR + m];
        const int     crow = cbase + ctile * 16 + hi * 8;

        if (nsel == 0) {
            v8f a0 = tile_gemm<0>(Fh, Wh + 0);
            v8f a1 = tile_gemm<16>(Fh, Wh + 16);
#pragma unroll
            for (int r = 0; r < 8; ++r) {
                float* o = out + ((size_t)b * NC + (size_t)(crow + r)) * NP;
                o[0 + m]  = a0[r];
                o[16 + m] = a1[r];
            }
        } else {
            v8f a2 = tile_gemm<32>(Fh, Wh + 32);
            v8f a3 = tile_gemm<48>(Fh, Wh + 48);
#pragma unroll
            for (int r = 0; r < 8; ++r) {
                float* o = out + ((size_t)b * NC + (size_t)(crow + r)) * NP;
                o[32 + m] = a2[r];
                if (m == 0) o[48] = a3[r];   // only p=48 is a real position
            }
        }
        __syncthreads();
    }
}

extern "C" void kernel_launch(void* const* d_in, const int* in_sizes, int n_in,
                              void* d_out, int out_size, void* d_ws, size_t ws_size,
                              hipStream_t stream) {
    const float* feats = (const float*)d_in[0];
    const float* boxes = (const float*)d_in[1];
    const int*   imH   = (const int*)d_in[2];
    const int*   imW   = (const int*)d_in[3];
    float*       outp  = (float*)d_out;

    const int B = in_sizes[1] / 4; // boxes are (B, 4)
    roi_bilinear_wmma_kernel<<<B, 256, 0, stream>>>(feats, boxes, imH, imW, outp);
    (void)d_ws; (void)ws_size; (void)n_in; (void)out_size;
}